// FALCON_46205258170442
// MI455X (gfx1250) — compile-verified
//
#include <hip/hip_runtime.h>
#include <math.h>

// ---------------------------------------------------------------------------
// FALCON ObjectSomeValuesFrom forward, MI455X / gfx1250 (wave32, WMMA).
//
//   N = 8192 entities (8000 named + 192 anon), D = 128.
//   Phase A (WMMA GEMV): rowdot[i] = e_all[i] . w_l ; colw[i] = e_all[i] . w_r
//   Phase B (tiny):      cdot = c_emb . w_l ; rdot = r_emb . w_l
//   Phase C (LDS max):   out[i] = max_j sig(rowdot[i]+rdot + colw[j]+b)
//                                     * sig(cdot + colw[j]+b)
// ---------------------------------------------------------------------------

#define N_NAMED 8000
#define ANON_E  192
#define N_ALL   (N_NAMED + ANON_E)   // 8192
#define EMB_DIM 128
#define N_TILES (N_ALL / 16)         // 512 (8000 is a multiple of 16 -> no straddle)

typedef __attribute__((ext_vector_type(2))) float v2f;
typedef __attribute__((ext_vector_type(8))) float v8f;

// ---------------------------------------------------------------------------
// Phase A: one wave32 per 16-row tile of e_all; accumulate 32 x (K=4) WMMA
// steps of V_WMMA_F32_16X16X4_F32.  B has w_l in column 0, w_r in column 1,
// zeros elsewhere, so D[:,0] = tile . w_l and D[:,1] = tile . w_r.
//
// A-matrix 16x4 f32 layout (ISA 7.12.2): lane L holds row M=L&15,
//   VGPR g holds K = (L>>4)*2 + g.  B 4x16 is the transpose-symmetric layout:
//   lane L holds column N=L&15, VGPR g holds K = (L>>4)*2 + g.
// D 16x16 f32: VGPR g, lanes 0-15 -> (M=g, N=lane); lanes 16-31 -> (M=8+g).
// ---------------------------------------------------------------------------
__global__ __launch_bounds__(256) void falcon_gemv_wmma(
    const float* __restrict__ e_table,     // [8000,128]
    const float* __restrict__ anon_e_emb,  // [192,128]
    const float* __restrict__ fc0_w,       // [256] = [w_l | w_r]
    float* __restrict__ ws_rowdot,         // [8192] out: e_all . w_l
    float* __restrict__ ws_colw)           // [8192] out: e_all . w_r
{
    const int tid  = threadIdx.x;
    const int lane = tid & 31;
    const int tile = blockIdx.x * (blockDim.x >> 5) + (tid >> 5);
    if (tile >= N_TILES) return;

    const int row  = tile * 16 + (lane & 15);
    const int nCol = lane & 15;           // B column this lane feeds
    const int koff = (lane >> 4) * 2;     // K sub-offset for this lane half

    const float* src = (row < N_NAMED)
                         ? (e_table    + (size_t)row * EMB_DIM)
                         : (anon_e_emb + (size_t)(row - N_NAMED) * EMB_DIM);
    // lane nCol==0 feeds w_l, nCol==1 feeds w_r, others feed zeros
    const float* wsrc = fc0_w + nCol * EMB_DIM;

    v8f acc = {};
#pragma unroll
    for (int kb = 0; kb < EMB_DIM / 4; ++kb) {
        const int k = kb * 4 + koff;
        float2 av = *(const float2*)(src + k);
        float2 bv = make_float2(0.f, 0.f);
        if (nCol < 2) bv = *(const float2*)(wsrc + k);
        v2f a; a[0] = av.x; a[1] = av.y;
        v2f b; b[0] = bv.x; b[1] = bv.y;
        acc = __builtin_amdgcn_wmma_f32_16x16x4_f32(
            /*neg_a=*/false, a, /*neg_b=*/false, b,
            /*c_mod=*/(short)0, acc, /*reuse_a=*/false, /*reuse_b=*/false);
    }

    // Extract columns 0 (rowdot) and 1 (colw): lanes {0,16} and {1,17}.
    const int mbase = (lane >> 4) * 8;
    if (nCol < 2) {
        float* dst = (nCol == 0) ? ws_rowdot : ws_colw;
#pragma unroll
        for (int g = 0; g < 8; ++g)
            dst[tile * 16 + mbase + g] = acc[g];
    }
}

// ---------------------------------------------------------------------------
// Phase B: cdot = c_table[c_id] . w_l ; rdot = r_table[r_id] . w_l.
// One block of 128 threads, deterministic fixed-tree LDS reduction.
// ---------------------------------------------------------------------------
__global__ __launch_bounds__(128) void falcon_scalar_dots(
    const float* __restrict__ c_table, const float* __restrict__ r_table,
    const float* __restrict__ fc0_w,
    const int* __restrict__ c_id, const int* __restrict__ r_id,
    float* __restrict__ ws_scal)   // [2] : {cdot, rdot}
{
    __shared__ float rc[EMB_DIM];
    __shared__ float rr[EMB_DIM];
    const int t  = threadIdx.x;
    const int ci = c_id[0];
    const int ri = r_id[0];
    const float wl = fc0_w[t];                      // w_l[t]
    rc[t] = c_table[(size_t)ci * EMB_DIM + t] * wl;
    rr[t] = r_table[(size_t)ri * EMB_DIM + t] * wl;
    __syncthreads();
#pragma unroll
    for (int s = EMB_DIM / 2; s > 0; s >>= 1) {
        if (t < s) { rc[t] += rc[t + s]; rr[t] += rr[t + s]; }
        __syncthreads();
    }
    if (t == 0) { ws_scal[0] = rc[0]; ws_scal[1] = rr[0]; }
}

// ---------------------------------------------------------------------------
// Phase C: the 8192^2 sigmoid/mul/max sweep.  Both j-indexed vectors
// (colw[j]+b and c_fs[j]) live entirely in 64 KB of LDS (320 KB available
// per WGP).  256 blocks x 256 threads; 8 threads per row, j strided by 8
// (consecutive DWORD addresses across lanes -> conflict-free), wave-local
// shfl_xor max reduction over the 8 sub-lanes (wave32).
// ---------------------------------------------------------------------------
__global__ __launch_bounds__(256) void falcon_max_sweep(
    const float* __restrict__ ws_rowdot,
    const float* __restrict__ ws_colw,
    const float* __restrict__ ws_scal,
    const float* __restrict__ fc0_b,
    float* __restrict__ out)
{
    __shared__ float s_sb[N_ALL];   // colw[j] + b
    __shared__ float s_cf[N_ALL];   // c_fs[j] = sigmoid(cdot + colw[j] + b)

    const int tid  = threadIdx.x;
    const float b    = fc0_b[0];
    const float cdot = ws_scal[0];
    const float rdot = ws_scal[1];

    for (int j = tid; j < N_ALL; j += blockDim.x) {
        const float sb = ws_colw[j] + b;
        s_sb[j] = sb;
        s_cf[j] = 1.0f / (1.0f + __expf(-(cdot + sb)));
    }
    __syncthreads();

    const int r   = blockIdx.x * 32 + (tid >> 3);   // row handled by this thread
    const int sub = tid & 7;                        // j-stride phase
    const float rowv = ws_rowdot[r] + rdot;

    float m = -INFINITY;
#pragma unroll 4
    for (int j = sub; j < N_ALL; j += 8) {
        const float t   = rowv + s_sb[j];
        const float sig = 1.0f / (1.0f + __expf(-t));
        m = fmaxf(m, sig * s_cf[j]);
    }
    // reduce over the 8 consecutive lanes that share this row (wave32)
    m = fmaxf(m, __shfl_xor(m, 1, 32));
    m = fmaxf(m, __shfl_xor(m, 2, 32));
    m = fmaxf(m, __shfl_xor(m, 4, 32));
    if (sub == 0) out[r] = m;
}

// ---------------------------------------------------------------------------
// Workspace layout (floats): [0,8192) rowdot | [8192,16384) colw | [16384,+2) scalars
// ---------------------------------------------------------------------------
extern "C" void kernel_launch(void* const* d_in, const int* in_sizes, int n_in,
                              void* d_out, int out_size, void* d_ws, size_t ws_size,
                              hipStream_t stream) {
    const float* anon_e_emb = (const float*)d_in[0];
    const float* e_table    = (const float*)d_in[1];
    const float* c_table    = (const float*)d_in[2];
    const float* r_table    = (const float*)d_in[3];
    const float* fc0_w      = (const float*)d_in[4];
    const float* fc0_b      = (const float*)d_in[5];
    const int*   c_id       = (const int*)d_in[6];
    const int*   r_id       = (const int*)d_in[7];
    float*       out        = (float*)d_out;

    float* ws_rowdot = (float*)d_ws;
    float* ws_colw   = ws_rowdot + N_ALL;
    float* ws_scal   = ws_rowdot + 2 * N_ALL;

    // Phase A: 512 tiles, 8 waves (256 threads) per block -> 64 blocks.
    falcon_gemv_wmma<<<dim3(N_TILES / 8), dim3(256), 0, stream>>>(
        e_table, anon_e_emb, fc0_w, ws_rowdot, ws_colw);

    // Phase B: single tiny block.
    falcon_scalar_dots<<<dim3(1), dim3(128), 0, stream>>>(
        c_table, r_table, fc0_w, c_id, r_id, ws_scal);

    // Phase C: 8192 rows / 32 rows-per-block -> 256 blocks.
    falcon_max_sweep<<<dim3(N_ALL / 32), dim3(256), 0, stream>>>(
        ws_rowdot, ws_colw, ws_scal, fc0_b, out);
}